// GMMLoss_37477884625584
// MI455X (gfx1250) — compile-verified
//
#include <hip/hip_runtime.h>
#include <math.h>

typedef __attribute__((ext_vector_type(2))) float v2f;
typedef __attribute__((ext_vector_type(8))) float v8f;

#define N_SAMP   262144            // B*T*C = 32*1024*8
#define DIM      8
#define KMIX     8
#define FEAT     64
#define NFTOT    (N_SAMP * FEAT)   // 16,777,216
#define NCHUNK   (N_SAMP / 4)      // 65,536 four-sample chunks

// workspace layout (float offsets)
#define WS_SUMSQ   0
#define WS_ENERGY  1
#define WS_COVDIAG 2
#define WS_GSUM    8      // [8]
#define WS_M1      16     // [8*8]
#define WS_S2      80     // [8*64]
#define WS_COEF    592    // [8]
#define WS_MU      600    // [8*8]
#define WS_INV     664    // [8*8*8]
#define WS_TOTAL   1176

union V8u { v8f v; float f[8]; };

// ---------------------------------------------------------------- reduction
__device__ __forceinline__ float blockReduce256(float v, float* sh8, int tid) {
#pragma unroll
    for (int off = 16; off > 0; off >>= 1) v += __shfl_down(v, off, 32);
    if ((tid & 31) == 0) sh8[tid >> 5] = v;
    __syncthreads();
    float s = 0.0f;
    if (tid == 0) {
#pragma unroll
        for (int i = 0; i < 8; ++i) s += sh8[i];
    }
    return s;
}

// ---------------------------------------------------------------- kernel 1
__global__ __launch_bounds__(256) void recon_kernel(const float* __restrict__ x,
                                                    const float* __restrict__ xh,
                                                    float* __restrict__ ws) {
    __shared__ float sh8[8];
    const int tid = threadIdx.x;
    float acc = 0.0f;
    const float4* x4  = (const float4*)x;
    const float4* xh4 = (const float4*)xh;
    const int n4 = NFTOT / 4;
    for (int i = blockIdx.x * 256 + tid; i < n4; i += gridDim.x * 256) {
        float4 a = x4[i];
        float4 b = xh4[i];
        float dx = b.x - a.x, dy = b.y - a.y, dz = b.z - a.z, dw = b.w - a.w;
        acc += dx * dx + dy * dy + dz * dz + dw * dw;
    }
    float tot = blockReduce256(acc, sh8, tid);
    if (tid == 0) atomicAdd(&ws[WS_SUMSQ], tot);
}

// ---------------------------------------------------------------- kernel 2
// gamma^T * [z | 1]  and  gamma^T * (z_i*z_j)  via V_WMMA_F32_16X16X4_F32.
// Each wave consumes 4 samples/iteration: A = gamma (M=k rows padded to 16,
// K=sample), B tiles hold z columns (+ ones column for gsum) and z_i*z_j.
__global__ __launch_bounds__(256) void moments_kernel(const float* __restrict__ z,
                                                      const float* __restrict__ gamma,
                                                      float* __restrict__ ws) {
    const int tid  = threadIdx.x;
    const int lane = tid & 31;
    const int wave = tid >> 5;
    const int gw   = blockIdx.x * 8 + wave;
    const int nw   = gridDim.x * 8;

    const int sub = lane >> 4;   // selects WMMA-K rows {0,1} vs {2,3}
    const int m   = lane & 15;   // A row index / B column index
    const int mm  = m & 7;
    const int s0  = 2 * sub;     // first sample handled by this half-wave
    const int s1  = 2 * sub + 1;

    v8f c_m1 = {};
    v8f c_s2_0 = {}, c_s2_1 = {}, c_s2_2 = {}, c_s2_3 = {};

    for (int c = gw; c < NCHUNK; c += nw) {
        const int base = c * 32;
        // coalesced: lane s*8+j holds gamma[4c+s][j] / z[4c+s][j]
        float gv = gamma[base + lane];
        float zv = z[base + lane];

        // A tile: A[m][s] = gamma[sample s][component m], rows 8..15 = 0
        v2f a;
        float a0 = __shfl(gv, s0 * 8 + mm, 32);
        float a1 = __shfl(gv, s1 * 8 + mm, 32);
        a.x = (m < 8) ? a0 : 0.0f;
        a.y = (m < 8) ? a1 : 0.0f;

        // B tile for M1 + gsum: col d<8 -> z[s][d]; col 8 -> 1; else 0
        float z0 = __shfl(zv, s0 * 8 + mm, 32);
        float z1 = __shfl(zv, s1 * 8 + mm, 32);
        v2f b;
        b.x = (m < 8) ? z0 : ((m == 8) ? 1.0f : 0.0f);
        b.y = (m < 8) ? z1 : ((m == 8) ? 1.0f : 0.0f);
        c_m1 = __builtin_amdgcn_wmma_f32_16x16x4_f32(false, a, false, b,
                                                     (short)0, c_m1, false, false);

        // B tiles for S2: column g = 16*t + m maps to (i,j) = (g>>3, g&7)
#pragma unroll
        for (int t = 0; t < 4; ++t) {
            const int g = t * 16 + m;
            const int i = g >> 3, j = g & 7;
            float zi0 = __shfl(zv, s0 * 8 + i, 32);
            float zj0 = __shfl(zv, s0 * 8 + j, 32);
            float zi1 = __shfl(zv, s1 * 8 + i, 32);
            float zj1 = __shfl(zv, s1 * 8 + j, 32);
            v2f bs;
            bs.x = zi0 * zj0;
            bs.y = zi1 * zj1;
            if (t == 0)      c_s2_0 = __builtin_amdgcn_wmma_f32_16x16x4_f32(false, a, false, bs, (short)0, c_s2_0, false, false);
            else if (t == 1) c_s2_1 = __builtin_amdgcn_wmma_f32_16x16x4_f32(false, a, false, bs, (short)0, c_s2_1, false, false);
            else if (t == 2) c_s2_2 = __builtin_amdgcn_wmma_f32_16x16x4_f32(false, a, false, bs, (short)0, c_s2_2, false, false);
            else             c_s2_3 = __builtin_amdgcn_wmma_f32_16x16x4_f32(false, a, false, bs, (short)0, c_s2_3, false, false);
        }
    }

    // cross-wave reduction in LDS; meaningful D rows M=0..7 live in
    // VGPR r=0..7 at lanes 0..15 (N = lane)
    __shared__ float red[5 * 8 * 8 * 16];   // [acc][wave][row][col] = 20 KB
    V8u ua[5];
    ua[0].v = c_m1;
    ua[1].v = c_s2_0; ua[2].v = c_s2_1; ua[3].v = c_s2_2; ua[4].v = c_s2_3;
    if (lane < 16) {
#pragma unroll
        for (int acc = 0; acc < 5; ++acc)
#pragma unroll
            for (int r = 0; r < 8; ++r)
                red[((acc * 8 + wave) * 8 + r) * 16 + lane] = ua[acc].f[r];
    }
    __syncthreads();

    for (int e = tid; e < 5 * 8 * 16; e += 256) {
        const int acc = e >> 7;
        const int rem = e & 127;
        const int r = rem >> 4;
        const int n = rem & 15;
        float s = 0.0f;
#pragma unroll
        for (int w = 0; w < 8; ++w) s += red[((acc * 8 + w) * 8 + r) * 16 + n];
        if (acc == 0) {
            if (n < 8)       atomicAdd(&ws[WS_M1 + r * 8 + n], s);
            else if (n == 8) atomicAdd(&ws[WS_GSUM + r], s);
        } else {
            const int g = (acc - 1) * 16 + n;
            atomicAdd(&ws[WS_S2 + r * 64 + g], s);
        }
    }
}

// ---------------------------------------------------------------- kernel 3
// one wave; lane k owns mixture component k: phi, mu, cov(+eps I),
// Gauss-Jordan inverse, Cholesky of 2*pi*cov, coef = phi / sqrt(prod diag L)
__global__ void stats_kernel(float* __restrict__ ws) {
    __shared__ float covS[8][8][8];
    __shared__ float invS[8][8][8];
    __shared__ float chol[8][8][8];
    const int k = threadIdx.x;
    if (k >= 8) return;

    const float gs  = ws[WS_GSUM + k];
    const float phi = gs / (float)N_SAMP;

    float mu[8];
#pragma unroll
    for (int d = 0; d < 8; ++d) {
        mu[d] = ws[WS_M1 + k * 8 + d] / gs;
        ws[WS_MU + k * 8 + d] = mu[d];
    }

    float cd = 0.0f;
    for (int i = 0; i < 8; ++i)
        for (int j = 0; j < 8; ++j) {
            float cv = ws[WS_S2 + k * 64 + i * 8 + j] / gs - mu[i] * mu[j];
            if (i == j) cv += 1e-12f;
            covS[k][i][j] = cv;
            chol[k][i][j] = 6.283185307179586f * cv;
            invS[k][i][j] = (i == j) ? 1.0f : 0.0f;
            if (i == j) cd += 1.0f / cv;
        }
    atomicAdd(&ws[WS_COVDIAG], cd);

    // Gauss-Jordan (SPD, no pivoting)
    for (int p = 0; p < 8; ++p) {
        const float ip = 1.0f / covS[k][p][p];
        for (int j = 0; j < 8; ++j) { covS[k][p][j] *= ip; invS[k][p][j] *= ip; }
        for (int i = 0; i < 8; ++i) {
            if (i == p) continue;
            const float f = covS[k][i][p];
            for (int j = 0; j < 8; ++j) {
                covS[k][i][j] -= f * covS[k][p][j];
                invS[k][i][j] -= f * invS[k][p][j];
            }
        }
    }
    for (int i = 0; i < 8; ++i)
        for (int j = 0; j < 8; ++j)
            ws[WS_INV + k * 64 + i * 8 + j] = invS[k][i][j];

    // Cholesky of 2*pi*cov; det_cov = prod diag(L)
    float det = 1.0f;
    for (int i = 0; i < 8; ++i) {
        for (int j = 0; j <= i; ++j) {
            float s = chol[k][i][j];
            for (int t = 0; t < j; ++t) s -= chol[k][i][t] * chol[k][j][t];
            chol[k][i][j] = (i == j) ? sqrtf(s) : s / chol[k][j][j];
        }
        det *= chol[k][i][i];
    }
    ws[WS_COEF + k] = phi / sqrtf(det);
}

// ---------------------------------------------------------------- kernel 4
__global__ __launch_bounds__(256) void energy_kernel(const float* __restrict__ z,
                                                     float* __restrict__ ws) {
    __shared__ float s_inv[512];
    __shared__ float s_mu[64];
    __shared__ float s_coef[8];
    __shared__ float sh8[8];
    const int tid = threadIdx.x;
    for (int i = tid; i < 512; i += 256) s_inv[i] = ws[WS_INV + i];
    if (tid < 64) s_mu[tid] = ws[WS_MU + tid];
    if (tid < 8)  s_coef[tid] = ws[WS_COEF + tid];
    __syncthreads();

    float acc = 0.0f;
    for (int n = blockIdx.x * 256 + tid; n < N_SAMP; n += gridDim.x * 256) {
        const float4* zp = (const float4*)(z + (size_t)n * 8);
        float4 za = zp[0], zb = zp[1];
        float zr[8] = {za.x, za.y, za.z, za.w, zb.x, zb.y, zb.z, zb.w};
        float mix = 0.0f;
#pragma unroll
        for (int k = 0; k < 8; ++k) {
            float d[8];
#pragma unroll
            for (int i = 0; i < 8; ++i) d[i] = zr[i] - s_mu[k * 8 + i];
            float quad = 0.0f;
#pragma unroll
            for (int i = 0; i < 8; ++i) {
                float ri = 0.0f;
#pragma unroll
                for (int j = 0; j < 8; ++j) ri += s_inv[k * 64 + i * 8 + j] * d[j];
                quad += d[i] * ri;
            }
            mix += s_coef[k] * __expf(-0.5f * quad);
        }
        acc += -__logf(mix + 1e-12f);
    }
    float tot = blockReduce256(acc, sh8, tid);
    if (tid == 0) atomicAdd(&ws[WS_ENERGY], tot);
}

// ---------------------------------------------------------------- kernel 5
__global__ void finalize_kernel(const float* __restrict__ ws, float* __restrict__ out) {
    if (threadIdx.x == 0 && blockIdx.x == 0) {
        const float recon  = ws[WS_SUMSQ] / (float)NFTOT;
        const float energy = ws[WS_ENERGY] / (float)N_SAMP;
        out[0] = recon + 0.1f * energy + 0.005f * ws[WS_COVDIAG];
    }
}

// ---------------------------------------------------------------- launch
extern "C" void kernel_launch(void* const* d_in, const int* in_sizes, int n_in,
                              void* d_out, int out_size, void* d_ws, size_t ws_size,
                              hipStream_t stream) {
    const float* x     = (const float*)d_in[0];
    const float* x_hat = (const float*)d_in[1];
    const float* z     = (const float*)d_in[2];
    const float* gamma = (const float*)d_in[3];
    float* ws  = (float*)d_ws;
    float* out = (float*)d_out;

    hipMemsetAsync(d_ws, 0, WS_TOTAL * sizeof(float), stream);

    recon_kernel<<<1024, 256, 0, stream>>>(x, x_hat, ws);
    moments_kernel<<<128, 256, 0, stream>>>(z, gamma, ws);
    stats_kernel<<<1, 32, 0, stream>>>(ws);
    energy_kernel<<<512, 256, 0, stream>>>(z, ws);
    finalize_kernel<<<1, 1, 0, stream>>>(ws, out);
}